// RankedListLoss_17884243820951
// MI455X (gfx1250) — compile-verified
//
#include <hip/hip_runtime.h>
#include <hip/hip_bf16.h>
#include <math.h>

#define BDIM 512
#define MARGIN 0.4f

typedef __attribute__((ext_vector_type(2))) float v2f;
typedef __attribute__((ext_vector_type(8))) float v8f;

// ---------------------------------------------------------------------------
// Kernel 1: L2-normalize rows. One wave32 per row, 8 rows per 256-thread block.
// Also computes sq[row] = sum(x_norm^2) exactly as the reference does.
// ---------------------------------------------------------------------------
__global__ __launch_bounds__(256) void rll_normalize(const float* __restrict__ emb,
                                                     float* __restrict__ x,
                                                     float* __restrict__ sq) {
  const int row  = blockIdx.x * 8 + (threadIdx.x >> 5);
  const int lane = threadIdx.x & 31;
  const float4* src = (const float4*)(emb + row * BDIM);
  float4 v[4];
  float ss = 0.0f;
#pragma unroll
  for (int i = 0; i < 4; ++i) {
    v[i] = src[lane + 32 * i];
    ss += v[i].x * v[i].x + v[i].y * v[i].y + v[i].z * v[i].z + v[i].w * v[i].w;
  }
#pragma unroll
  for (int off = 16; off > 0; off >>= 1) ss += __shfl_xor(ss, off, 32);
  const float inv = 1.0f / fmaxf(sqrtf(ss), 1e-12f);
  float4* dst = (float4*)(x + row * BDIM);
  float ssn = 0.0f;
#pragma unroll
  for (int i = 0; i < 4; ++i) {
    float4 w = v[i];
    w.x *= inv; w.y *= inv; w.z *= inv; w.w *= inv;
    dst[lane + 32 * i] = w;
    ssn += w.x * w.x + w.y * w.y + w.z * w.z + w.w * w.w;
  }
#pragma unroll
  for (int off = 16; off > 0; off >>= 1) ssn += __shfl_xor(ssn, off, 32);
  if (lane == 0) sq[row] = ssn;
}

// ---------------------------------------------------------------------------
// Kernel 2: G = x * x^T via V_WMMA_F32_16X16X4_F32, register-blocked: one wave
// computes a 16x64 output strip (4 N-tiles), reusing the A fragment across
// 4 WMMAs per K-step (5 loads : 4 wmma instead of 2 : 1).
// Fused epilogue: dist = sqrt(max(sq_i + sq_j - 2*g, 0)).
// ---------------------------------------------------------------------------
__global__ __launch_bounds__(32) void rll_gemm_dist(const float* __restrict__ x,
                                                    const float* __restrict__ sq,
                                                    float* __restrict__ dist) {
  const int bid  = blockIdx.x;              // 0..255
  const int tm   = (bid & 31) << 4;         // row-tile origin (16 rows)
  const int tnb  = (bid >> 5) << 6;         // col-strip origin (64 cols)
  const int lane = threadIdx.x;             // wave32
  const int half = lane >> 4;               // 0: K={0,1}, 1: K={2,3}
  const int idx  = lane & 15;
  const int koff = half << 1;

  const float* arow = x + (tm + idx) * BDIM + koff;         // A: M=idx
  const float* brow = x + (tnb + idx) * BDIM + koff;        // B tile 0: N=idx

  v8f acc[4] = {};
#pragma unroll 4
  for (int k = 0; k < BDIM; k += 4) {
    const v2f a = *(const v2f*)(arow + k);
    v2f b[4];
#pragma unroll
    for (int t = 0; t < 4; ++t)
      b[t] = *(const v2f*)(brow + t * (16 * BDIM) + k);
#pragma unroll
    for (int t = 0; t < 4; ++t)
      // 8 args: (neg_a, A, neg_b, B, c_mod, C, reuse_a, reuse_b)
      acc[t] = __builtin_amdgcn_wmma_f32_16x16x4_f32(false, a, false, b[t],
                                                     (short)0, acc[t], false, false);
  }

#pragma unroll
  for (int t = 0; t < 4; ++t) {
    const int n = tnb + 16 * t + idx;
    const float sqn = sq[n];
#pragma unroll
    for (int v = 0; v < 8; ++v) {
      const int m = tm + v + (half << 3);   // C/D: M = vgpr + 8*half
      const float d2 = sq[m] + sqn - 2.0f * acc[t][v];
      dist[m * BDIM + n] = sqrtf(fmaxf(d2, 0.0f));
    }
  }
}

// ---------------------------------------------------------------------------
// Kernel 3: per-row violation counting, block-cooperative. One 256-thread
// block per row i. Each thread owns 2 k-slots (dist + neg flag in registers);
// the block iterates positives j in fixed order (uniform LDS predicate, no
// divergence) and tree-reduces an integer count -> deterministic.
// ---------------------------------------------------------------------------
__global__ __launch_bounds__(256) void rll_count(const float* __restrict__ dist,
                                                 const int* __restrict__ labels,
                                                 float* __restrict__ partials) {
  __shared__ float drow[BDIM];
  __shared__ unsigned char posf[BDIM];
  __shared__ int wred[8];
  const int i  = blockIdx.x;
  const int t  = threadIdx.x;
  const int li = labels[i];

  int neg0, neg1;
  for (int j = t; j < BDIM; j += 256) {
    drow[j] = dist[i * BDIM + j];
    const int lj = labels[j];
    const int isneg = (lj != li) ? 1 : 0;
    posf[j] = (unsigned char)((lj == li) && (j != i));
    if (j == t) neg0 = isneg; else neg1 = isneg;
  }
  __syncthreads();

  // Hoist this thread's two k-slots into registers.
  const float d0 = drow[t];
  const float d1 = drow[t + 256];

  float total = 0.0f;
  for (int j = 0; j < BDIM; ++j) {
    if (!posf[j]) continue;                 // uniform across the block
    const float thr = drow[j] + MARGIN;     // LDS broadcast
    int c = ((neg0 && (d0 < thr)) ? 1 : 0) + ((neg1 && (d1 < thr)) ? 1 : 0);
#pragma unroll
    for (int off = 16; off > 0; off >>= 1) c += __shfl_xor(c, off, 32);
    if ((t & 31) == 0) wred[t >> 5] = c;
    __syncthreads();
    if (t == 0) {
      int ct = 0;
#pragma unroll
      for (int w = 0; w < 8; ++w) ct += wred[w];
      if (ct > 0) total += log1pf((float)ct);   // * ALPHA (=1.0)
    }
    __syncthreads();
  }
  if (t == 0) partials[i] = total;
}

// ---------------------------------------------------------------------------
// Kernel 4: deterministic final reduction + scale by 1/(n + 1e-8).
// ---------------------------------------------------------------------------
__global__ __launch_bounds__(256) void rll_reduce(const float* __restrict__ partials,
                                                  float* __restrict__ out) {
  __shared__ float s[256];
  const int t = threadIdx.x;
  s[t] = partials[t] + partials[t + 256];
  __syncthreads();
#pragma unroll
  for (int off = 128; off > 0; off >>= 1) {
    if (t < off) s[t] += s[t + off];
    __syncthreads();
  }
  if (t == 0) out[0] = s[0] / (512.0f + 1e-8f);
}

// ---------------------------------------------------------------------------
extern "C" void kernel_launch(void* const* d_in, const int* in_sizes, int n_in,
                              void* d_out, int out_size, void* d_ws, size_t ws_size,
                              hipStream_t stream) {
  const float* emb    = (const float*)d_in[0];   // (512, 512) f32
  const int*   labels = (const int*)d_in[1];     // (512,) i32
  float* out = (float*)d_out;                    // scalar f32

  float* x        = (float*)d_ws;                // 512*512 f32
  float* dist     = x + BDIM * BDIM;             // 512*512 f32
  float* sq       = dist + BDIM * BDIM;          // 512 f32
  float* partials = sq + BDIM;                   // 512 f32

  rll_normalize<<<BDIM / 8, 256, 0, stream>>>(emb, x, sq);
  rll_gemm_dist<<<(BDIM / 16) * (BDIM / 64), 32, 0, stream>>>(x, sq, dist);
  rll_count<<<BDIM, 256, 0, stream>>>(dist, labels, partials);
  rll_reduce<<<1, 256, 0, stream>>>(partials, out);
}